// LNSA_loss_47193100649164
// MI455X (gfx1250) — compile-verified
//
#include <hip/hip_runtime.h>
#include <hip/hip_bf16.h>
#include <math.h>
#include <stdint.h>

#define N_PTS 8192
#define DX    512
#define DZ    64
#define KNN   5
#define EPSF  1e-7f

#define BM 64
#define BN 32
#define LSTR 516   // LDS row stride in floats: 2064B (16B aligned), bank-conflict-free

typedef __attribute__((ext_vector_type(2))) float v2f;
typedef __attribute__((ext_vector_type(8))) float v8f;
typedef int v4i __attribute__((vector_size(16)));

// ---------------- async global->LDS copy of 16B (ASYNCcnt-tracked), with fallback
__device__ __forceinline__ void async_copy16(const float* g, float* l) {
#if __has_builtin(__builtin_amdgcn_global_load_async_to_lds_b128)
    __builtin_amdgcn_global_load_async_to_lds_b128(
        (__attribute__((address_space(1))) v4i*)g,
        (__attribute__((address_space(3))) v4i*)l,
        0, 0);
#else
    unsigned lds = (unsigned)(uintptr_t)l;
    unsigned long long ga = (unsigned long long)(uintptr_t)g;
    asm volatile("global_load_async_to_lds_b128 %0, %1, off"
                 :: "v"(lds), "v"(ga) : "memory");
#endif
}

__device__ __forceinline__ void wait_async0() {
#if __has_builtin(__builtin_amdgcn_s_wait_asynccnt)
    __builtin_amdgcn_s_wait_asynccnt(0);
#else
    asm volatile("s_wait_asynccnt 0" ::: "memory");
#endif
}

// ---------------------------------------------------------------- column means
__global__ void colmean_kernel(const float* __restrict__ A, float* __restrict__ mu,
                               int n, int d) {
    int c = blockIdx.x * blockDim.x + threadIdx.x;
    if (c >= d) return;
    float s = 0.f;
    for (int r = 0; r < n; ++r) s += A[(size_t)r * d + c];
    mu[c] = s / (float)n;
}

// ------------------------------------------- per-row sq-norm and centered norm
__global__ void rowstats_kernel(const float* __restrict__ A, const float* __restrict__ mu,
                                float* __restrict__ sq, float* __restrict__ cn, int d) {
    int lane = threadIdx.x & 31;
    int wr   = threadIdx.x >> 5;
    int row  = blockIdx.x * 8 + wr;
    float s = 0.f, c2 = 0.f;
    for (int j = lane; j < d; j += 32) {
        float x = A[(size_t)row * d + j];
        s += x * x;
        float c = x - mu[j];
        c2 += c * c;
    }
    for (int o = 16; o > 0; o >>= 1) {
        s  += __shfl_down(s,  o, 32);
        c2 += __shfl_down(c2, o, 32);
    }
    if (lane == 0) { sq[row] = s; cn[row] = sqrtf(c2); }
}

// ------------------------------ exact 0.98 quantile via rank selection (N=8192)
__global__ void quantile98_kernel(const float* __restrict__ cnX,
                                  const float* __restrict__ cnZ,
                                  float* __restrict__ normA) {
    __shared__ float s[N_PTS];
    __shared__ float qlo, qhi;
    const float* src = (blockIdx.x == 0) ? cnX : cnZ;
    for (int i = threadIdx.x; i < N_PTS; i += blockDim.x) s[i] = src[i];
    __syncthreads();
    const int LO = 8027, HI = 8028;           // floor/ceil of 0.98*(8192-1)
    for (int i = threadIdx.x; i < N_PTS; i += blockDim.x) {
        float v = s[i];
        int rank = 0;
        for (int j = 0; j < N_PTS; ++j) {
            float u = s[j];
            rank += (u < v) || (u == v && j < i);
        }
        if (rank == LO) qlo = v;
        if (rank == HI) qhi = v;
    }
    __syncthreads();
    if (threadIdx.x == 0) {
        float w = 0.98f * 8191.0f - 8027.0f;
        normA[blockIdx.x] = qlo * (1.0f - w) + qhi * w;
    }
}

// ------- fused Gram (WMMA f32) -> distance -> streaming top-5 -> lid_X + nn idx
// Double-buffered B panels staged via GLOBAL_LOAD_ASYNC_TO_LDS_B128 so staging
// of tile nt+1 overlaps the 128-WMMA K-loop on tile nt (only 1 WG/WGP at this
// LDS footprint, so there is no other workgroup to hide staging latency).
__global__ void gram_topk_kernel(const float* __restrict__ X,
                                 const float* __restrict__ sqX,
                                 const float* __restrict__ normA,
                                 float* __restrict__ lidX,
                                 int*   __restrict__ nnidx) {
    extern __shared__ char smem[];
    float* sA    = (float*)smem;                 // BM x LSTR
    float* sB0   = sA  + BM * LSTR;              // BN x LSTR (buffer 0)
    float* sB1   = sB0 + BN * LSTR;              // BN x LSTR (buffer 1)
    float* dtile = sB1 + BN * LSTR;              // BM x 33
    float* mval  = dtile + BM * 33;              // BM x 20
    int*   midx  = (int*)(mval + BM * 20);       // BM x 20

    const int t    = threadIdx.x;
    const int row0 = blockIdx.x * BM;
    const int wave = t >> 5;
    const int lane = t & 31;
    const int wm = wave >> 1;        // 0..3: 16-row stripe
    const int wn = wave & 1;         // 0..1: 16-col stripe
    const int lo = lane & 15;
    const int hi = lane >> 4;        // 0/1
    const int kb = hi * 2;           // K sub-offset per ISA A/B layout

    // stage the 64x512 A row-panel once (async, 16B granules)
    for (int it = t; it < BM * (DX / 4); it += blockDim.x) {
        int r = it >> 7, c4 = it & 127;
        async_copy16(X + (size_t)(row0 + r) * DX + c4 * 4, &sA[r * LSTR + c4 * 4]);
    }
    // stage B panel 0
    for (int it = t; it < BN * (DX / 4); it += blockDim.x) {
        int r = it >> 7, c4 = it & 127;
        async_copy16(X + (size_t)r * DX + c4 * 4, &sB0[r * LSTR + c4 * 4]);
    }

    float rsq[8];
    #pragma unroll
    for (int v = 0; v < 8; ++v) rsq[v] = sqX[row0 + wm * 16 + v + 8 * hi];

    // private top-5 (ascending, index tie-break) — 4 threads per row
    float t5v[5]; int t5i[5];
    #pragma unroll
    for (int i = 0; i < 5; ++i) { t5v[i] = 3.4e38f; t5i[i] = 0x7fffffff; }
    const int prow = t >> 2, psub = t & 3;

    wait_async0();
    __syncthreads();

    for (int nt = 0; nt < N_PTS / BN; ++nt) {
        const int j0 = nt * BN;
        float* bufC = (nt & 1) ? sB1 : sB0;   // compute buffer
        float* bufN = (nt & 1) ? sB0 : sB1;   // next-staging buffer

        // kick off async staging of the NEXT B panel (overlaps the K-loop)
        if (nt + 1 < N_PTS / BN) {
            const int jn = j0 + BN;
            for (int it = t; it < BN * (DX / 4); it += blockDim.x) {
                int r = it >> 7, c4 = it & 127;
                async_copy16(X + (size_t)(jn + r) * DX + c4 * 4,
                             &bufN[r * LSTR + c4 * 4]);
            }
            // light prefetch of the panel after that into cache
            if (nt + 2 < N_PTS / BN) {
                const float* nb = X + (size_t)(jn + BN) * DX;
                for (int it = t; it < BN * 8; it += blockDim.x)
                    __builtin_prefetch(nb + (it >> 3) * DX + (it & 7) * 64, 0, 1);
            }
        }

        v8f acc = {0.f, 0.f, 0.f, 0.f, 0.f, 0.f, 0.f, 0.f};
        const float* aBase = &sA[(wm * 16 + lo) * LSTR + kb];
        const float* bBase = &bufC[(wn * 16 + lo) * LSTR + kb];
        #pragma unroll 8
        for (int k = 0; k < DX; k += 4) {
            v2f a = *(const v2f*)(aBase + k);
            v2f b = *(const v2f*)(bBase + k);
            acc = __builtin_amdgcn_wmma_f32_16x16x4_f32(
                false, a, false, b, (short)0, acc, false, false);
        }

        // squared distances (clamped like the reference) into LDS
        float csq = sqX[j0 + wn * 16 + lo];
        #pragma unroll
        for (int v = 0; v < 8; ++v) {
            float d2 = rsq[v] + csq - 2.0f * acc[v];
            dtile[(wm * 16 + v + 8 * hi) * 33 + (wn * 16 + lo)] = fmaxf(d2, 0.f);
        }
        __syncthreads();

        // streaming top-5 partials
        #pragma unroll
        for (int q = 0; q < 8; ++q) {
            int c = psub * 8 + q;
            float dv = dtile[prow * 33 + c];
            int ji = j0 + c;
            if (dv < t5v[4] || (dv == t5v[4] && ji < t5i[4])) {
                t5v[4] = dv; t5i[4] = ji;
                #pragma unroll
                for (int p = 4; p > 0; --p) {
                    bool sw = (t5v[p] < t5v[p-1]) ||
                              (t5v[p] == t5v[p-1] && t5i[p] < t5i[p-1]);
                    if (sw) {
                        float tv = t5v[p]; t5v[p] = t5v[p-1]; t5v[p-1] = tv;
                        int   ti = t5i[p]; t5i[p] = t5i[p-1]; t5i[p-1] = ti;
                    }
                }
            }
        }
        // next buffer must be fully written (own waves' async cnt + all waves)
        wait_async0();
        __syncthreads();
    }

    // merge 4 partial top-5 lists per row
    #pragma unroll
    for (int i = 0; i < 5; ++i) {
        mval[prow * 20 + psub * 5 + i] = t5v[i];
        midx[prow * 20 + psub * 5 + i] = t5i[i];
    }
    __syncthreads();

    if (t < BM) {
        float cv[20]; int ci[20];
        for (int i = 0; i < 20; ++i) { cv[i] = mval[t * 20 + i]; ci[i] = midx[t * 20 + i]; }
        float selv[5]; int seli[5];
        for (int s = 0; s < 5; ++s) {
            int best = -1;
            for (int i = 0; i < 20; ++i) {
                if (ci[i] < 0) continue;
                if (best < 0 || cv[i] < cv[best] ||
                    (cv[i] == cv[best] && ci[i] < ci[best])) best = i;
            }
            selv[s] = cv[best]; seli[s] = ci[best]; ci[best] = -1;
        }
        // rank 0 is self; neighbors are ranks 1..4 (same as idx[:,1:])
        float na = normA[0];
        float lg[4]; float sum = 0.f;
        for (int s = 1; s < 5; ++s) {
            float val = (sqrtf(selv[s]) + EPSF) / na;
            lg[s - 1] = log10f(val);
            sum += lg[s - 1];
            nnidx[(size_t)(row0 + t) * 4 + (s - 1)] = seli[s];
        }
        lidX[row0 + t] = (sum - 4.0f * lg[3]) / (float)KNN + EPSF;
    }
}

// ------------------------------------------------ zero the scalar output
__global__ void zero_out_kernel(float* out) {
    if (threadIdx.x == 0 && blockIdx.x == 0) out[0] = 0.f;
}

// --------------- Z-side: distances only at the 4 X-neighbors, loss accumulation
__global__ void loss_z_kernel(const float* __restrict__ Zm,
                              const float* __restrict__ sqZ,
                              const float* __restrict__ normA,
                              const float* __restrict__ lidX,
                              const int*   __restrict__ nnidx,
                              float* __restrict__ out) {
    int lane = threadIdx.x & 31;
    int wr   = threadIdx.x >> 5;
    int i    = blockIdx.x * 8 + wr;
    float z0 = Zm[(size_t)i * DZ + lane];
    float z1 = Zm[(size_t)i * DZ + lane + 32];
    float nb = normA[1];
    float lg[4]; float sum = 0.f;
    for (int s = 0; s < 4; ++s) {
        int j = nnidx[(size_t)i * 4 + s];
        float dot = z0 * Zm[(size_t)j * DZ + lane] + z1 * Zm[(size_t)j * DZ + lane + 32];
        for (int o = 16; o > 0; o >>= 1) dot += __shfl_down(dot, o, 32);
        if (lane == 0) {
            float d2 = sqZ[i] + sqZ[j] - 2.0f * dot;
            d2 = fmaxf(d2, 0.f);
            float ev = (sqrtf(d2) + EPSF) / nb;
            lg[s] = log10f(ev);
            sum += lg[s];
        }
    }
    if (lane == 0) {
        float lidZ = (sum - 4.0f * lg[3]) / (float)KNN + EPSF;
        float d = lidX[i] - lidZ;
        atomicAdd(out, d * d / (float)N_PTS);
    }
}

extern "C" void kernel_launch(void* const* d_in, const int* in_sizes, int n_in,
                              void* d_out, int out_size, void* d_ws, size_t ws_size,
                              hipStream_t stream) {
    const float* X = (const float*)d_in[0];   // 8192 x 512
    const float* Z = (const float*)d_in[1];   // 8192 x 64
    float* out = (float*)d_out;

    float* w     = (float*)d_ws;
    float* muX   = w;                          // 512
    float* muZ   = w + 512;                    // 64
    float* sqX   = w + 1024;                   // 8192
    float* cnX   = sqX + N_PTS;                // 8192
    float* sqZ   = cnX + N_PTS;                // 8192
    float* cnZ   = sqZ + N_PTS;                // 8192
    float* normA = cnZ + N_PTS;                // 2 (+pad)
    float* lidX  = normA + 16;                 // 8192
    int*   nnidx = (int*)(lidX + N_PTS);       // 8192*4 ints

    colmean_kernel<<<dim3(2),    dim3(256), 0, stream>>>(X, muX, N_PTS, DX);
    colmean_kernel<<<dim3(1),    dim3(64),  0, stream>>>(Z, muZ, N_PTS, DZ);
    rowstats_kernel<<<dim3(1024), dim3(256), 0, stream>>>(X, muX, sqX, cnX, DX);
    rowstats_kernel<<<dim3(1024), dim3(256), 0, stream>>>(Z, muZ, sqZ, cnZ, DZ);
    quantile98_kernel<<<dim3(2), dim3(1024), 0, stream>>>(cnX, cnZ, normA);

    size_t lds_bytes = (size_t)(BM * LSTR + 2 * BN * LSTR + BM * 33 + BM * 20) * 4
                     + (size_t)(BM * 20) * 4;   // ~276 KB, fits 320 KB WGP LDS
    gram_topk_kernel<<<dim3(N_PTS / BM), dim3(256), lds_bytes, stream>>>(
        X, sqX, normA, lidX, nnidx);

    zero_out_kernel<<<dim3(1), dim3(32), 0, stream>>>(out);
    loss_z_kernel<<<dim3(N_PTS / 8), dim3(256), 0, stream>>>(
        Z, sqZ, normA, lidX, nnidx, out);
}